// NonlearnableCell_12610023981641
// MI455X (gfx1250) — compile-verified
//
#include <hip/hip_runtime.h>
#include <math.h>

#define BB 64
#define TT 4096
#define II 64
#define HH 64
#define GG 256
#define LN_EPS_F 1e-5f

typedef __attribute__((ext_vector_type(2))) float v2f;
typedef __attribute__((ext_vector_type(8))) float v8f;

// ---------------------------------------------------------------------------
// Phase 1: ih[b,t,:] = LayerNorm(xs[b,t,:] @ w_ih[b] + b_ih[b])  (parallel in t)
// f32 WMMA 16x16x4 path; one workgroup = 16 timesteps x 256 gates of one batch.
// ---------------------------------------------------------------------------
__global__ __launch_bounds__(256) void lstm_phase1(
        const float* __restrict__ xs, const float* __restrict__ w_ih,
        const float* __restrict__ b_ih, float* __restrict__ ih_out)
{
    const int b  = blockIdx.y;
    const int t0 = blockIdx.x * 16;

    __shared__ float Alds[16][II + 1];   // padded: conflict-free column reads
    __shared__ float Cl[16][GG + 4];

    const int tid   = threadIdx.x;       // 0..255
    const int wave  = tid >> 5;
    const int lane  = tid & 31;
    const int nhalf = lane >> 4;         // 0/1 : which K-half of the fragment
    const int nlan  = lane & 15;         // row (A) / col (B) within tile

    // Stage the shared 16x64 x-tile once for all 8 waves.
    for (int idx = tid; idx < 16 * II; idx += 256) {
        const int r = idx >> 6, c = idx & 63;
        Alds[r][c] = xs[((size_t)b * TT + t0 + r) * II + c];
    }
    __syncthreads();

    // Each wave produces 2 of the 16 16x16 N-tiles.
    for (int nt = wave; nt < 16; nt += 8) {
        const int n0 = nt * 16;

        // Preload this wave's 64x16 w_ih block into registers (reused K-loop).
        // B layout (4x16 f32): lane = n + 16*(k>>1), vgpr = k&1.
        float Breg[32];
#pragma unroll
        for (int kk = 0; kk < 16; ++kk) {
#pragma unroll
            for (int j = 0; j < 2; ++j) {
                const int k = 4 * kk + 2 * nhalf + j;
                Breg[2 * kk + j] = w_ih[((size_t)b * II + k) * GG + n0 + nlan];
            }
        }

        v8f acc = {0.f, 0.f, 0.f, 0.f, 0.f, 0.f, 0.f, 0.f};
#pragma unroll
        for (int kk = 0; kk < 16; ++kk) {
            const int kb = 4 * kk + 2 * nhalf;
            v2f a, bf;
            // A layout (16x4 f32): lanes 0-15 M=0..15 K={0,1}; lanes 16-31 K={2,3}.
            a.x = Alds[nlan][kb + 0];
            a.y = Alds[nlan][kb + 1];
            bf.x = Breg[2 * kk + 0];
            bf.y = Breg[2 * kk + 1];
            acc = __builtin_amdgcn_wmma_f32_16x16x4_f32(
                false, a, false, bf, (short)0, acc, false, false);
        }
        // D layout: vgpr r -> lanes 0-15: (M=r, N=lane); lanes 16-31: (M=8+r).
#pragma unroll
        for (int r = 0; r < 8; ++r)
            Cl[r + 8 * nhalf][n0 + nlan] = acc[r];
    }
    __syncthreads();

    // Bias + LayerNorm per row; wave w owns rows 2w, 2w+1.
#pragma unroll
    for (int rr = 0; rr < 2; ++rr) {
        const int r = wave * 2 + rr;
        float s = 0.f, s2 = 0.f;
#pragma unroll
        for (int c = lane; c < GG; c += 32) {
            const float v = Cl[r][c] + b_ih[(size_t)b * GG + c];
            Cl[r][c] = v;
            s += v; s2 += v * v;
        }
#pragma unroll
        for (int off = 16; off; off >>= 1) {
            s  += __shfl_xor(s,  off, 32);
            s2 += __shfl_xor(s2, off, 32);
        }
        const float mean = s * (1.f / GG);
        const float var  = s2 * (1.f / GG) - mean * mean;
        const float inv  = rsqrtf(var + LN_EPS_F);
#pragma unroll
        for (int c = lane; c < GG; c += 32)
            ih_out[((size_t)b * TT + t0 + r) * GG + c] = (Cl[r][c] - mean) * inv;
    }
}

// ---------------------------------------------------------------------------
// Phase 2: sequential recurrence. One workgroup (256 threads) per batch chain.
// w_hh column held in 64 registers per lane; h broadcast from LDS.
// ---------------------------------------------------------------------------
__global__ __launch_bounds__(256) void lstm_phase2(
        const float* __restrict__ ih, const float* __restrict__ h0,
        const float* __restrict__ c0, const float* __restrict__ w_hh,
        const float* __restrict__ b_hh, float* __restrict__ ys,
        float* __restrict__ hy, float* __restrict__ cy)
{
    const int b    = blockIdx.x;
    const int j    = threadIdx.x;        // gate index 0..255
    const int wave = j >> 5, lane = j & 31;

    __shared__ float hbuf[HH];
    __shared__ float gbuf[GG];
    __shared__ float red[16];
    __shared__ float cred[4];

    float Wreg[HH];                      // w_hh[b][:, j] resident in VGPRs
#pragma unroll
    for (int k = 0; k < HH; ++k)
        Wreg[k] = w_hh[((size_t)b * HH + k) * GG + j];
    const float bh = b_hh[(size_t)b * GG + j];

    if (j < HH) hbuf[j] = h0[(size_t)b * HH + j];
    float c = (j < HH) ? c0[(size_t)b * HH + j] : 0.f;
    float hylast = 0.f;
    __syncthreads();

    for (int t = 0; t < TT; ++t) {
        // Independent of h: issue early so latency hides under the matvec.
        const float ihv = ih[((size_t)b * TT + t) * GG + j];
        if (t + 1 < TT)
            __builtin_prefetch(&ih[((size_t)b * TT + t + 1) * GG + j], 0, 0);

        // hh_pre[j] = b_hh[j] + sum_k h[k] * W[k][j]  (LDS broadcast + FMA)
        float acc = bh;
#pragma unroll
        for (int k = 0; k < HH; ++k)
            acc += hbuf[k] * Wreg[k];

        // LayerNorm over the 256 gates (workgroup-wide two-level reduction).
        float s = acc, s2 = acc * acc;
#pragma unroll
        for (int off = 16; off; off >>= 1) {
            s  += __shfl_xor(s,  off, 32);
            s2 += __shfl_xor(s2, off, 32);
        }
        if (lane == 0) { red[wave] = s; red[8 + wave] = s2; }
        __syncthreads();
        float ts = 0.f, ts2 = 0.f;
#pragma unroll
        for (int w = 0; w < 8; ++w) { ts += red[w]; ts2 += red[8 + w]; }
        const float mean = ts * (1.f / GG);
        const float var  = ts2 * (1.f / GG) - mean * mean;
        const float hh   = (acc - mean) * rsqrtf(var + LN_EPS_F);

        const float gate = ihv + hh;
        const float act  = (j >= 2 * HH && j < 3 * HH)
                               ? tanhf(gate)
                               : 1.f / (1.f + __expf(-gate));
        gbuf[j] = act;
        __syncthreads();

        // Cell update + LayerNorm over 64 cell units (threads 0..63 = waves 0,1).
        float cp = 0.f;
        if (j < HH) {
            cp = gbuf[HH + j] * c + gbuf[j] * gbuf[2 * HH + j];
            float cs = cp, cs2 = cp * cp;
#pragma unroll
            for (int off = 16; off; off >>= 1) {
                cs  += __shfl_xor(cs,  off, 32);
                cs2 += __shfl_xor(cs2, off, 32);
            }
            if (lane == 0) { cred[wave] = cs; cred[2 + wave] = cs2; }
        }
        __syncthreads();
        if (j < HH) {
            const float cm = (cred[0] + cred[1]) * (1.f / HH);
            const float cv = (cred[2] + cred[3]) * (1.f / HH) - cm * cm;
            const float cn = (cp - cm) * rsqrtf(cv + LN_EPS_F);
            c = cn;
            hylast = gbuf[3 * HH + j] * tanhf(cn);
            hbuf[j] = hylast;
            ys[((size_t)b * TT + t) * HH + j] = hylast;
        }
        __syncthreads();   // h ready for next step; red/gbuf free for reuse
    }

    if (j < HH) {
        hy[(size_t)b * HH + j] = hylast;
        cy[(size_t)b * HH + j] = c;
    }
}

// ---------------------------------------------------------------------------
extern "C" void kernel_launch(void* const* d_in, const int* in_sizes, int n_in,
                              void* d_out, int out_size, void* d_ws, size_t ws_size,
                              hipStream_t stream) {
    (void)in_sizes; (void)n_in; (void)out_size; (void)ws_size;
    const float* xs   = (const float*)d_in[0];
    const float* h0   = (const float*)d_in[1];
    const float* c0   = (const float*)d_in[2];
    const float* w_ih = (const float*)d_in[3];
    const float* w_hh = (const float*)d_in[4];
    const float* b_ih = (const float*)d_in[5];
    const float* b_hh = (const float*)d_in[6];

    float* out = (float*)d_out;
    float* ys  = out;                                  // (B,T,H)
    float* hy  = out + (size_t)BB * TT * HH;           // (B,1,H)
    float* cy  = hy + (size_t)BB * HH;                 // (B,1,H)

    float* ihbuf = (float*)d_ws;                       // B*T*G f32 = 256 MB

    dim3 g1(TT / 16, BB);
    lstm_phase1<<<g1, 256, 0, stream>>>(xs, w_ih, b_ih, ihbuf);
    lstm_phase2<<<BB, 256, 0, stream>>>(ihbuf, h0, c0, w_hh, b_hh, ys, hy, cy);
}